// ARC_22711787061507
// MI455X (gfx1250) — compile-verified
//
#include <hip/hip_runtime.h>
#include <math.h>

#define BATCH   64
#define CHN     3
#define IMG     256
#define GS      16
#define CO      512
#define IN_DIM  768   // 16*16*3
#define NGATES  2048  // 4*CO
#define STEPS   16    // 2 * num_glimpses (8)
#define FB_EPS  1e-4f
#define PI_F    3.14159265358979323846f

typedef __attribute__((ext_vector_type(2))) float v2f;
typedef __attribute__((ext_vector_type(8))) float v8f;

// CDNA5 fp32 WMMA: D(16x16,f32) = A(16x4,f32) * B(4x16,f32) + C
__device__ __forceinline__ v8f wmma4(v2f a, v2f b, v8f c) {
    return __builtin_amdgcn_wmma_f32_16x16x4_f32(
        /*neg_a=*/false, a, /*neg_b=*/false, b,
        /*c_mod=*/(short)0, c, /*reuse_a=*/false, /*reuse_b=*/false);
}

__device__ __forceinline__ float sigmoidf_(float x) { return 1.0f / (1.0f + expf(-x)); }

// ---------------------------------------------------------------------------
// Zero h and c state (contiguous region of n floats)
// ---------------------------------------------------------------------------
__global__ void k_init(float* __restrict__ p, int n) {
    int i = blockIdx.x * blockDim.x + threadIdx.x;
    if (i < n) p[i] = 0.0f;
}

// ---------------------------------------------------------------------------
// Per batch: gp = tanh(h @ Wg^T + bg), then build normalized filterbanks
// Fh[b,g,i] (center=gp0) and Fw[b,g,i] (center=gp1), delta=gp2. 1 block / batch.
// ---------------------------------------------------------------------------
__global__ __launch_bounds__(256) void k_fb(const float* __restrict__ hstate,
                                            const float* __restrict__ Wg,
                                            const float* __restrict__ bg,
                                            float* __restrict__ Fh,
                                            float* __restrict__ Fw) {
    const int b = blockIdx.x;
    const int t = threadIdx.x;            // 256 threads
    __shared__ float sred[256];
    __shared__ float sgp[3];
    __shared__ float fx[GS * IMG];
    __shared__ float rowsum[GS];

    // gp[j] = tanh(sum_k h[b,k]*Wg[j,k] + bg[j]), co=512
    for (int j = 0; j < 3; ++j) {
        float p = hstate[b * CO + t]       * Wg[j * CO + t]
                + hstate[b * CO + t + 256] * Wg[j * CO + t + 256];
        sred[t] = p;
        __syncthreads();
        for (int s = 128; s > 0; s >>= 1) {
            if (t < s) sred[t] += sred[t + s];
            __syncthreads();
        }
        if (t == 0) sgp[j] = tanhf(sred[0] + bg[j]);
        __syncthreads();
    }

    const float delta  = sgp[2];
    const float ad     = fabsf(delta);
    const float deltas = ((float)IMG / (float)GS) * (1.0f - ad);
    const float gamma  = expf(1.0f - 2.0f * ad);
    const int   wv = t >> 5, ln = t & 31;

    for (int which = 0; which < 2; ++which) {
        const float center = sgp[which];
        const float cpix   = (IMG - 1.0f) * (center + 1.0f) * 0.5f;
        // each thread handles image column i = t, all 16 glimpse rows
        for (int g = 0; g < GS; ++g) {
            float mu = cpix + deltas * ((float)g - (GS - 1.0f) * 0.5f);
            float d  = ((float)t - mu) / gamma;
            fx[g * IMG + t] = 1.0f / (PI_F * gamma * (1.0f + d * d));
        }
        __syncthreads();
        // row sums: wave wv handles rows 2*wv, 2*wv+1 (8 waves x 2 = 16 rows)
        for (int gi = 0; gi < 2; ++gi) {
            int g = wv * 2 + gi;
            float s = 0.0f;
            for (int i = ln; i < IMG; i += 32) s += fx[g * IMG + i];
            for (int off = 16; off > 0; off >>= 1) s += __shfl_down(s, off, 32);
            if (ln == 0) rowsum[g] = s;
        }
        __syncthreads();
        float* dst = (which == 0) ? Fh : Fw;
        for (int g = 0; g < GS; ++g)
            dst[((size_t)b * GS + g) * IMG + t] = fx[g * IMG + t] / (rowsum[g] + FB_EPS);
        __syncthreads();
    }
}

// ---------------------------------------------------------------------------
// Glimpse extraction: out16x16[b,c] = Fh[b](16x256) @ img[b,c](256x256) @ Fw[b]^T
// One block per (c,b); 4 waves. Stage 1: each wave owns 64 consecutive image
// columns (4 N-tiles) -> A fragment loaded once per K-step, 4 independent
// WMMAs issued back-to-back with 8 B-loads in flight. Stage 2 K-split across
// waves with LDS partial reduction. Writes x[b, c*256 + g*16 + k].
// ---------------------------------------------------------------------------
__global__ __launch_bounds__(128) void k_glimpse(const float* __restrict__ img_pairs,
                                                 int turn,
                                                 const float* __restrict__ Fh,
                                                 const float* __restrict__ Fw,
                                                 float* __restrict__ xbuf) {
    const int c = blockIdx.x, b = blockIdx.y;
    const int t = threadIdx.x;            // 128 threads = 4 waves
    const int wv = t >> 5, ln = t & 31;
    const int half = ln >> 4;             // 0: K rows +0/1 ; 1: K rows +2/3
    const int l16  = ln & 15;

    const float* img = img_pairs + ((((size_t)b * 2 + (turn & 1)) * CHN + c) * IMG) * IMG;
    const float* FhB = Fh + (size_t)b * GS * IMG;
    const float* FwB = Fw + (size_t)b * GS * IMG;

    __shared__ float T[GS * IMG];    // 16 KB: stage-1 result (16 x 256)
    __shared__ float P[4 * GS * GS]; // per-wave stage-2 partials

    // ---- stage 1: T = Fh @ img ; wave wv owns columns [wv*64, wv*64+63]
    {
        const int n0 = wv * 64;
        v8f acc0 = {}, acc1 = {}, acc2 = {}, acc3 = {};
        for (int k0 = 0; k0 < IMG; k0 += 4) {
            const int ka = k0 + half * 2;
            // shared A fragment for all 4 N tiles
            v2f a;  a.x = FhB[l16 * IMG + ka];
                    a.y = FhB[l16 * IMG + ka + 1];
            const float* r0 = img + (size_t)ka * IMG + n0 + l16;
            const float* r1 = r0 + IMG;
            v2f b0; b0.x = r0[0];   b0.y = r1[0];
            v2f b1; b1.x = r0[16];  b1.y = r1[16];
            v2f b2; b2.x = r0[32];  b2.y = r1[32];
            v2f b3; b3.x = r0[48];  b3.y = r1[48];
            // speculative warm of rows ~32 ahead (unconditional, no branches)
            __builtin_prefetch(r0 + 32 * IMG, 0, 3);
            acc0 = wmma4(a, b0, acc0);
            acc1 = wmma4(a, b1, acc1);
            acc2 = wmma4(a, b2, acc2);
            acc3 = wmma4(a, b3, acc3);
        }
        #pragma unroll
        for (int v = 0; v < 8; ++v) {
            const int m = v + half * 8;
            T[m * IMG + n0 +  0 + l16] = acc0[v];
            T[m * IMG + n0 + 16 + l16] = acc1[v];
            T[m * IMG + n0 + 32 + l16] = acc2[v];
            T[m * IMG + n0 + 48 + l16] = acc3[v];
        }
    }
    __syncthreads();

    // ---- stage 2: out = T @ Fw^T ; B[k][n] = Fw[b][n][k] ; K split over waves
    v8f acc = {};
    for (int k0 = wv * 64; k0 < wv * 64 + 64; k0 += 4) {
        const int ka = k0 + half * 2;
        v2f a;  a.x  = T[l16 * IMG + ka];
                a.y  = T[l16 * IMG + ka + 1];
        v2f bb; bb.x = FwB[l16 * IMG + ka];
                bb.y = FwB[l16 * IMG + ka + 1];
        acc = wmma4(a, bb, acc);
    }
    #pragma unroll
    for (int v = 0; v < 8; ++v) {
        const int m = v + half * 8;
        P[wv * 256 + m * 16 + l16] = acc[v];
    }
    __syncthreads();

    for (int e = t; e < 256; e += 128) {
        float s = P[e] + P[256 + e] + P[512 + e] + P[768 + e];
        xbuf[(size_t)b * IN_DIM + c * 256 + e] = s;   // e = g*16 + k
    }
}

// ---------------------------------------------------------------------------
// gates(64x2048) = x(64x768) @ W_ih^T + h(64x512) @ W_hh^T + b_ih + b_hh
// grid (4 Mtiles, 8); 8 waves/block; each wave owns TWO 16-wide N tiles so the
// A fragment is loaded once per K-step and two independent WMMAs pipeline.
// ---------------------------------------------------------------------------
__global__ __launch_bounds__(256) void k_gates(const float* __restrict__ xbuf,
                                               const float* __restrict__ hstate,
                                               const float* __restrict__ W_ih,
                                               const float* __restrict__ W_hh,
                                               const float* __restrict__ b_ih,
                                               const float* __restrict__ b_hh,
                                               float* __restrict__ gates) {
    const int t = threadIdx.x;
    const int wv = t >> 5, ln = t & 31;
    const int half = ln >> 4, l16 = ln & 15;
    const int m0 = blockIdx.x * 16;
    const int nb = blockIdx.y * 256 + wv * 32;   // base of this wave's 2 tiles
    const int n0 = nb + l16;
    const int n1 = nb + 16 + l16;

    v8f acc0 = {}, acc1 = {};
    // x @ W_ih^T : B[k][n] = W_ih[n*768 + k]
    for (int k0 = 0; k0 < IN_DIM; k0 += 4) {
        const int ka = k0 + half * 2;
        v2f a;  a.x  = xbuf[(size_t)(m0 + l16) * IN_DIM + ka];
                a.y  = xbuf[(size_t)(m0 + l16) * IN_DIM + ka + 1];
        v2f b0; b0.x = W_ih[(size_t)n0 * IN_DIM + ka];
                b0.y = W_ih[(size_t)n0 * IN_DIM + ka + 1];
        v2f b1; b1.x = W_ih[(size_t)n1 * IN_DIM + ka];
                b1.y = W_ih[(size_t)n1 * IN_DIM + ka + 1];
        acc0 = wmma4(a, b0, acc0);
        acc1 = wmma4(a, b1, acc1);
    }
    // h @ W_hh^T
    for (int k0 = 0; k0 < CO; k0 += 4) {
        const int ka = k0 + half * 2;
        v2f a;  a.x  = hstate[(size_t)(m0 + l16) * CO + ka];
                a.y  = hstate[(size_t)(m0 + l16) * CO + ka + 1];
        v2f b0; b0.x = W_hh[(size_t)n0 * CO + ka];
                b0.y = W_hh[(size_t)n0 * CO + ka + 1];
        v2f b1; b1.x = W_hh[(size_t)n1 * CO + ka];
                b1.y = W_hh[(size_t)n1 * CO + ka + 1];
        acc0 = wmma4(a, b0, acc0);
        acc1 = wmma4(a, b1, acc1);
    }
    const float bias0 = b_ih[n0] + b_hh[n0];
    const float bias1 = b_ih[n1] + b_hh[n1];
    #pragma unroll
    for (int v = 0; v < 8; ++v) {
        const int m = m0 + v + half * 8;
        gates[(size_t)m * NGATES + n0] = acc0[v] + bias0;
        gates[(size_t)m * NGATES + n1] = acc1[v] + bias1;
    }
}

// ---------------------------------------------------------------------------
// LSTM pointwise update; writes h (and final output on last step)
// ---------------------------------------------------------------------------
__global__ void k_lstm(const float* __restrict__ gates,
                       float* __restrict__ hstate,
                       float* __restrict__ cstate,
                       float* __restrict__ final_out) {
    int idx = blockIdx.x * blockDim.x + threadIdx.x;
    if (idx >= BATCH * CO) return;
    const int b = idx >> 9, j = idx & (CO - 1);
    const float* gb = gates + (size_t)b * NGATES;
    float gi = gb[j], gf = gb[CO + j], gg = gb[2 * CO + j], go = gb[3 * CO + j];
    float cn = sigmoidf_(gf) * cstate[idx] + sigmoidf_(gi) * tanhf(gg);
    float hn = sigmoidf_(go) * tanhf(cn);
    cstate[idx] = cn;
    hstate[idx] = hn;
    if (final_out) final_out[idx] = hn;
}

// ---------------------------------------------------------------------------
extern "C" void kernel_launch(void* const* d_in, const int* in_sizes, int n_in,
                              void* d_out, int out_size, void* d_ws, size_t ws_size,
                              hipStream_t stream) {
    const float* img  = (const float*)d_in[0];
    const float* W_ih = (const float*)d_in[1];
    const float* W_hh = (const float*)d_in[2];
    const float* b_ih = (const float*)d_in[3];
    const float* b_hh = (const float*)d_in[4];
    const float* Wg   = (const float*)d_in[5];
    const float* bg   = (const float*)d_in[6];
    (void)in_sizes; (void)n_in; (void)out_size; (void)ws_size;

    float* ws    = (float*)d_ws;
    float* h     = ws;                                  // 64*512
    float* c     = h  + BATCH * CO;                     // 64*512
    float* Fh    = c  + BATCH * CO;                     // 64*16*256
    float* Fw    = Fh + BATCH * GS * IMG;               // 64*16*256
    float* xb    = Fw + BATCH * GS * IMG;               // 64*768
    float* gates = xb + BATCH * IN_DIM;                 // 64*2048

    const int state_n = 2 * BATCH * CO;                 // h and c contiguous
    k_init<<<(state_n + 255) / 256, 256, 0, stream>>>(h, state_n);

    for (int step = 0; step < STEPS; ++step) {
        k_fb<<<BATCH, 256, 0, stream>>>(h, Wg, bg, Fh, Fw);
        k_glimpse<<<dim3(CHN, BATCH), 128, 0, stream>>>(img, step, Fh, Fw, xb);
        k_gates<<<dim3(4, 8), 256, 0, stream>>>(xb, h, W_ih, W_hh, b_ih, b_hh, gates);
        k_lstm<<<(BATCH * CO + 255) / 256, 256, 0, stream>>>(
            gates, h, c, (step == STEPS - 1) ? (float*)d_out : nullptr);
    }
}